// CausalFieldAttentionV2_57629871178548
// MI455X (gfx1250) — compile-verified
//
#include <hip/hip_runtime.h>

typedef __attribute__((ext_vector_type(2))) float v2f;
typedef __attribute__((ext_vector_type(8))) float v8f;

#define D1    1024
#define NSEQ  8192
#define BATCH 4
#define GRIDF 512
#define HEADS 16
#define HD    64
#define TM    32
#define TN    128
#define KT    32
#define NTAP  56

// Matches reference: clip(int32(n/8191 * 511), 0, 511) in fp32.
__device__ __forceinline__ int field_cell(int n) {
    float s = ((float)n / 8191.0f) * 511.0f;
    int g = (int)s;
    if (g < 0) g = 0;
    if (g > 511) g = 511;
    return g;
}

__device__ __forceinline__ v8f wmma_f32(v2f a, v2f b, v8f c) {
    // V_WMMA_F32_16X16X4_F32 : D = A(16x4) x B(4x16) + C(16x16)
    return __builtin_amdgcn_wmma_f32_16x16x4_f32(false, a, false, b, (short)0, c,
                                                 false, false);
}

// CDNA5 async global -> LDS copy (16B per lane), tracked by ASYNCcnt.
// Generic pointer to __shared__ carries the LDS byte offset in its low 32 bits.
__device__ __forceinline__ void async_copy16(void* ldsDst, const void* gsrc) {
    asm volatile("global_load_async_to_lds_b128 %0, %1, off"
                 :: "v"((unsigned)(size_t)ldsDst), "v"(gsrc)
                 : "memory");
}
__device__ __forceinline__ void wait_async0() {
    asm volatile("s_wait_asynccnt 0x0" ::: "memory");
}

// ---------------------------------------------------------------------------
// Kernel 0: zero the field accumulator (B,G,D) = 8 MB
// ---------------------------------------------------------------------------
__global__ __launch_bounds__(256) void zero_field_kernel(float4* __restrict__ p) {
    p[(size_t)blockIdx.x * 256 + threadIdx.x] = float4{0.f, 0.f, 0.f, 0.f};
}

// ---------------------------------------------------------------------------
// Kernel 1: fused  [K|V] = x @ Wkv^T + b  ->  kmag = ||K_h||  ->
//           field[b, g(n), h*64+d] += V_h * kmag   (cell-reduced atomics)
// Block: 256 threads (8 waves), tile M=32 rows of x, N=128 = [K_h(64) | V_h(64)]
// grid = (H, B*N/32); h fastest so 16 consecutive blocks reuse the same x rows.
// Double-buffered LDS, async global->LDS copies overlap with WMMA.
// ---------------------------------------------------------------------------
__global__ __launch_bounds__(256) void kv_deposit_kernel(
    const float* __restrict__ x, const float* __restrict__ Wqkv,
    const float* __restrict__ bqkv, float* __restrict__ field)
{
    const int h       = blockIdx.x;
    const int rowBase = blockIdx.y * TM;          // global row in [0, B*N)
    const int tid     = threadIdx.x;
    const int lane    = tid & 31;
    const int wid     = tid >> 5;                 // 0..7
    const int wm      = wid & 1;                  // M half
    const int wn      = wid >> 1;                 // N quarter (0,1 = K; 2,3 = V)
    const int half    = lane >> 4;                // K-pair select within fragment
    const int l16     = lane & 15;

    __shared__ float4 As[2][KT / 4][TM];          // [buf][kstep][row]
    __shared__ float4 Bs[2][KT / 4][TN];          // [buf][kstep][col]
    __shared__ float  ksq[TM];                    // sum of squares -> kmag
    __shared__ float  dep[TM][HD + 1];            // V + bias (padded)
    __shared__ int    cellrow[TM];

    // per-thread staging coordinates
    const int ar  = tid & 31;                     // A row, A kstep = tid>>5
    const int aks = tid >> 5;
    const float* aSrc = x + (size_t)(rowBase + ar) * D1 + aks * 4;

    const int bn   = tid & 127;                   // B col (same for all 4 entries)
    const int bks0 = tid >> 7;                    // base kstep (0/1); entries at +2i
    const int wrow = (bn < HD) ? (D1 + h * HD + bn) : (2 * D1 + h * HD + (bn - HD));
    const float* bSrc = Wqkv + (size_t)wrow * D1 + bks0 * 4;

    v8f acc0 = {};
    v8f acc1 = {};

    auto issue = [&](int buf, int kf) {
        async_copy16(&As[buf][aks][ar], aSrc + kf);
        #pragma unroll
        for (int i = 0; i < 4; ++i)
            async_copy16(&Bs[buf][bks0 + 2 * i][bn], bSrc + kf + 8 * i);
    };

    issue(0, 0);
    for (int kk = 0; kk < D1; kk += KT) {
        const int buf = (kk >> 5) & 1;
        wait_async0();                            // my async copies into buf done
        __syncthreads();                          // everyone's copies done,
                                                  // everyone's reads of buf^1 done
        if (kk + KT < D1) issue(buf ^ 1, kk + KT);
        #pragma unroll
        for (int ks = 0; ks < KT / 4; ++ks) {
            v2f a  = ((const v2f*)&As[buf][ks][wm * 16 + l16])[half];
            v2f b0 = ((const v2f*)&Bs[buf][ks][wn * 32 + l16])[half];
            v2f b1 = ((const v2f*)&Bs[buf][ks][wn * 32 + 16 + l16])[half];
            acc0 = wmma_f32(a, b0, acc0);
            acc1 = wmma_f32(a, b1, acc1);
        }
    }

    const int nBase = rowBase & (NSEQ - 1);
    if (tid < TM) {
        ksq[tid]     = 0.f;
        cellrow[tid] = field_cell(nBase + tid);
    }
    __syncthreads();

    if (wn < 2) {
        // K columns: accumulate per-row sum of squares (with bias)
        float bk0 = bqkv[D1 + h * HD + wn * 32 + l16];
        float bk1 = bqkv[D1 + h * HD + wn * 32 + 16 + l16];
        #pragma unroll
        for (int r = 0; r < 8; ++r) {
            float v0 = acc0[r] + bk0;
            float v1 = acc1[r] + bk1;
            float s  = v0 * v0 + v1 * v1;
            s += __shfl_xor(s, 1);
            s += __shfl_xor(s, 2);
            s += __shfl_xor(s, 4);
            s += __shfl_xor(s, 8);                // reduce within each 16-lane half
            if (l16 == 0) atomicAdd(&ksq[wm * 16 + r + 8 * half], s);
        }
    } else {
        // V columns: stash V + bias into LDS tile
        int d0 = (wn - 2) * 32;
        float bv0 = bqkv[2 * D1 + h * HD + d0 + l16];
        float bv1 = bqkv[2 * D1 + h * HD + d0 + 16 + l16];
        #pragma unroll
        for (int r = 0; r < 8; ++r) {
            int row = wm * 16 + r + 8 * half;
            dep[row][d0 + l16]      = acc0[r] + bv0;
            dep[row][d0 + 16 + l16] = acc1[r] + bv1;
        }
    }
    __syncthreads();
    if (tid < TM) ksq[tid] = sqrtf(ksq[tid]);     // now holds kmag
    __syncthreads();

    // Cell-level pre-reduction: 32 rows span <= 3 grid cells; one atomic per
    // (cell, d) instead of per (row, d)  ->  ~12x fewer global atomics.
    {
        int d  = tid & (HD - 1);
        int j  = tid >> 6;                        // candidate cell 0..3
        int cj = cellrow[0] + j;
        int b  = rowBase >> 13;
        float sum = 0.f;
        bool  any = false;
        #pragma unroll
        for (int row = 0; row < TM; ++row) {
            if (cellrow[row] == cj) { sum += dep[row][d] * ksq[row]; any = true; }
        }
        if (any)
            atomicAdd(field + ((size_t)b * GRIDF + cj) * D1 + h * HD + d, sum);
    }
}

// ---------------------------------------------------------------------------
// Kernel 2: circular convolution along G (FFT-equivalent, 56 taps)
//   out[b,g,:] = sum_t w[t] * field[b, (g-256+t) & 511, :]
//   w[t] = exp(-2t) / (sum_i exp(-2i) + 1e-8)
// ---------------------------------------------------------------------------
__global__ __launch_bounds__(256) void conv_kernel(const float* __restrict__ fin,
                                                   float* __restrict__ fout)
{
    const int bg = blockIdx.x;                    // b*512 + g
    const int g  = bg & (GRIDF - 1);
    const int b  = bg >> 9;
    const int d  = threadIdx.x * 4;

    float z = 0.f;
    for (int i = 0; i < NTAP; ++i) z += expf(-2.0f * (float)i);
    const float invz = 1.0f / (z + 1e-8f);

    float4 acc = {0.f, 0.f, 0.f, 0.f};
    const float* base = fin + (size_t)b * GRIDF * D1;
    for (int t = 0; t < NTAP; ++t) {
        int gs  = (g - 256 + t) & (GRIDF - 1);
        float w = expf(-2.0f * (float)t) * invz;
        float4 f = *(const float4*)(base + (size_t)gs * D1 + d);
        acc.x = fmaf(w, f.x, acc.x);
        acc.y = fmaf(w, f.y, acc.y);
        acc.z = fmaf(w, f.z, acc.z);
        acc.w = fmaf(w, f.w, acc.w);
    }
    *(float4*)(fout + (size_t)bg * D1 + d) = acc;
}

// ---------------------------------------------------------------------------
// Kernel 3: out = gather(fieldc) @ Wout^T + bout
// A-row for token (b,n) is the contiguous row fieldc[b, g(n), :]  (gather fused
// into the async A copy). Same WMMA tiling / async double-buffering as kernel 1.
// ---------------------------------------------------------------------------
__global__ __launch_bounds__(256) void out_gemm_kernel(
    const float* __restrict__ fieldc, const float* __restrict__ Wout,
    const float* __restrict__ bout, float* __restrict__ out)
{
    const int colBase = blockIdx.x * TN;
    const int rowBase = blockIdx.y * TM;
    const int tid  = threadIdx.x;
    const int lane = tid & 31;
    const int wid  = tid >> 5;
    const int wm   = wid & 1;
    const int wn   = wid >> 1;
    const int half = lane >> 4;
    const int l16  = lane & 15;

    __shared__ float4 As[2][KT / 4][TM];
    __shared__ float4 Bs[2][KT / 4][TN];

    // per-thread staging coordinates (gather: row -> fieldc[b, g(n), :])
    const int ar  = tid & 31;
    const int aks = tid >> 5;
    const int rg  = rowBase + ar;
    const int rb  = rg >> 13;
    const int rn  = rg & (NSEQ - 1);
    const float* aSrc = fieldc + (size_t)(rb * GRIDF + field_cell(rn)) * D1 + aks * 4;

    const int bn   = tid & 127;
    const int bks0 = tid >> 7;
    const float* bSrc = Wout + (size_t)(colBase + bn) * D1 + bks0 * 4;

    v8f acc0 = {};
    v8f acc1 = {};

    auto issue = [&](int buf, int kf) {
        async_copy16(&As[buf][aks][ar], aSrc + kf);
        #pragma unroll
        for (int i = 0; i < 4; ++i)
            async_copy16(&Bs[buf][bks0 + 2 * i][bn], bSrc + kf + 8 * i);
    };

    issue(0, 0);
    for (int kk = 0; kk < D1; kk += KT) {
        const int buf = (kk >> 5) & 1;
        wait_async0();
        __syncthreads();
        if (kk + KT < D1) issue(buf ^ 1, kk + KT);
        #pragma unroll
        for (int ks = 0; ks < KT / 4; ++ks) {
            v2f a  = ((const v2f*)&As[buf][ks][wm * 16 + l16])[half];
            v2f b0 = ((const v2f*)&Bs[buf][ks][wn * 32 + l16])[half];
            v2f b1 = ((const v2f*)&Bs[buf][ks][wn * 32 + 16 + l16])[half];
            acc0 = wmma_f32(a, b0, acc0);
            acc1 = wmma_f32(a, b1, acc1);
        }
    }

    float bc0 = bout[colBase + wn * 32 + l16];
    float bc1 = bout[colBase + wn * 32 + 16 + l16];
    #pragma unroll
    for (int r = 0; r < 8; ++r) {
        size_t row = (size_t)(rowBase + wm * 16 + r + 8 * half);
        out[row * D1 + colBase + wn * 32 + l16]      = acc0[r] + bc0;
        out[row * D1 + colBase + wn * 32 + 16 + l16] = acc1[r] + bc1;
    }
}

// ---------------------------------------------------------------------------
extern "C" void kernel_launch(void* const* d_in, const int* in_sizes, int n_in,
                              void* d_out, int out_size, void* d_ws, size_t ws_size,
                              hipStream_t stream) {
    const float* x    = (const float*)d_in[0];   // (4, 8192, 1024)
    const float* Wqkv = (const float*)d_in[1];   // (3072, 1024)
    const float* bqkv = (const float*)d_in[2];   // (3072,)
    const float* Wout = (const float*)d_in[3];   // (1024, 1024)
    const float* bout = (const float*)d_in[4];   // (1024,)
    float* out = (float*)d_out;                  // (4, 8192, 1024)

    float* field  = (float*)d_ws;                                  // (B,G,D) 8MB
    float* fieldc = field + (size_t)BATCH * GRIDF * D1;            // (B,G,D) 8MB

    const int fieldVec4 = (BATCH * GRIDF * D1) / 4;                // 524288
    zero_field_kernel<<<fieldVec4 / 256, 256, 0, stream>>>((float4*)field);

    kv_deposit_kernel<<<dim3(HEADS, (BATCH * NSEQ) / TM), 256, 0, stream>>>(
        x, Wqkv, bqkv, field);

    conv_kernel<<<BATCH * GRIDF, 256, 0, stream>>>(field, fieldc);

    out_gemm_kernel<<<dim3(D1 / TN, (BATCH * NSEQ) / TM), 256, 0, stream>>>(
        fieldc, Wout, bout, out);
}